// VQVAE_77841987272820
// MI455X (gfx1250) — compile-verified
//
#include <hip/hip_runtime.h>
#include <hip/hip_bf16.h>

// VQ-VAE quantization for MI455X (gfx1250, wave32).
//   dist argmin via fp32 WMMA:  D = cb2[k] - 2 * (ze . cb[k])
//   latent = codebook[argmin], q_loss = 1.25 * mean((zq - ze)^2)

typedef __attribute__((ext_vector_type(2))) float v2f;
typedef __attribute__((ext_vector_type(8))) float v8f;

#define CDIM   128          // channels
#define KCB    8192         // codebook entries
#define NTOT   16384        // B*H*W
#define HW     1024         // H*W
#define MT     16           // rows per block
#define WAVES  4
#define COLTILES (KCB / 16) // 512
#define TILES_PER_WAVE (COLTILES / WAVES) // 128
#define APAD   132          // LDS row stride (floats), conflict-free

// ---------------- kernel 1: codebook squared norms ----------------
__global__ __launch_bounds__(256) void vq_cb_norms(const float* __restrict__ cb,
                                                   float* __restrict__ cb2) {
    int k = blockIdx.x * blockDim.x + threadIdx.x;
    if (k >= KCB) return;
    const float4* row = (const float4*)(cb + (size_t)k * CDIM);
    float s = 0.f;
#pragma unroll
    for (int i = 0; i < CDIM / 4; ++i) {
        float4 v = row[i];
        s += v.x * v.x + v.y * v.y + v.z * v.z + v.w * v.w;
    }
    cb2[k] = s;
}

// ---------------- kernel 2: WMMA distance + argmin + gather ----------------
__global__ __launch_bounds__(128) void vq_main(const float* __restrict__ ze,
                                               const float* __restrict__ cb,
                                               const float* __restrict__ cb2,
                                               float* __restrict__ latent,
                                               float* __restrict__ partial) {
    __shared__ float As[MT * APAD];      // ze tile, [m][c] padded
    __shared__ float dred[WAVES * MT];
    __shared__ int   kred[WAVES * MT];
    __shared__ int   idxs[MT];
    __shared__ float lred[WAVES];

    const int tid  = threadIdx.x;
    const int lane = tid & 31;
    const int wave = tid >> 5;
    const int half = lane >> 4;          // 0: rows 0-7 / 1: rows 8-15 (C/D layout)
    const int lr   = lane & 15;          // column-within-tile (C/D layout)

    const int n0   = blockIdx.x * MT;    // pixel-row tile base
    const int b    = n0 >> 10;           // batch (HW = 1024, tiles never cross)
    const int hw0  = n0 & (HW - 1);
    const size_t zbase = (size_t)b * (CDIM * HW) + hw0;

    // ---- stage ze tile into LDS: As[m*APAD + c] = z_e[b][c][hw0+m] ----
    {
        const int m  = tid & 15;
        const int c0 = tid >> 4;         // 0..7
#pragma unroll
        for (int i = 0; i < 16; ++i) {
            int c = c0 + i * 8;
            As[m * APAD + c] = ze[zbase + (size_t)c * HW + m];
        }
    }
    __syncthreads();

    // ---- preload A fragments for all 32 K-steps (V_WMMA_F32_16X16X4_F32 layout:
    //      lanes 0-15: {K=4s, 4s+1}, lanes 16-31: {K=4s+2, 4s+3}, M = lane%16) ----
    v2f afrag[32];
#pragma unroll
    for (int s = 0; s < 32; ++s) {
        int c = 4 * s + 2 * half;
        afrag[s] = *(const v2f*)&As[lr * APAD + c];   // ds_load_b64, 8B aligned
    }

    // ---- per-lane running argmin (slot r: row = r + 8*half, col = tile*16 + lr) ----
    float dmin[8];
    int   kmin[8];
#pragma unroll
    for (int r = 0; r < 8; ++r) { dmin[r] = 3.4e38f; kmin[r] = 0; }

    const int t0 = wave * TILES_PER_WAVE;
    for (int t = 0; t < TILES_PER_WAVE; ++t) {
        const int mycol = (t0 + t) * 16 + lr;
        const float* brow = cb + (size_t)mycol * CDIM + 2 * half;
        if (t + 1 < TILES_PER_WAVE)
            __builtin_prefetch((const void*)(brow + 16 * CDIM), 0, 1); // next tile row
        v8f acc = {};
#pragma unroll
        for (int s = 0; s < 32; ++s) {
            v2f bfrag = *(const v2f*)(brow + 4 * s);  // global_load_b64
            acc = __builtin_amdgcn_wmma_f32_16x16x4_f32(
                false, afrag[s], false, bfrag, (short)0, acc, false, false);
        }
        const float c2 = cb2[mycol];
#pragma unroll
        for (int r = 0; r < 8; ++r) {
            float d = c2 - 2.0f * acc[r];
            if (d < dmin[r]) { dmin[r] = d; kmin[r] = mycol; }
        }
    }

    // ---- cross-lane argmin within each 16-lane half (xor<16 stays in half) ----
#pragma unroll
    for (int off = 1; off < 16; off <<= 1) {
#pragma unroll
        for (int r = 0; r < 8; ++r) {
            float od = __shfl_xor(dmin[r], off, 32);
            int   ok = __shfl_xor(kmin[r], off, 32);
            if (od < dmin[r] || (od == dmin[r] && ok < kmin[r])) {
                dmin[r] = od; kmin[r] = ok;
            }
        }
    }
    if (lr == 0) {
#pragma unroll
        for (int r = 0; r < 8; ++r) {
            dred[wave * MT + half * 8 + r] = dmin[r];
            kred[wave * MT + half * 8 + r] = kmin[r];
        }
    }
    __syncthreads();

    // ---- combine across the 4 waves ----
    if (tid < MT) {
        float bd = dred[tid];
        int   bk = kred[tid];
#pragma unroll
        for (int w = 1; w < WAVES; ++w) {
            float od = dred[w * MT + tid];
            int   ok = kred[w * MT + tid];
            if (od < bd || (od == bd && ok < bk)) { bd = od; bk = ok; }
        }
        idxs[tid] = bk;
    }
    __syncthreads();

    // ---- gather codebook rows, write latent [B,C,H,W], accumulate loss ----
    float ls = 0.f;
    {
        const int m  = tid & 15;
        const int c0 = tid >> 4;
        const int k  = idxs[m];
        const float* crow = cb + (size_t)k * CDIM;
#pragma unroll
        for (int i = 0; i < 16; ++i) {
            int c = c0 + i * 8;
            float q = crow[c];
            float z = As[m * APAD + c];
            float e = q - z;
            ls += e * e;
            latent[zbase + (size_t)c * HW + m] = q;
        }
    }
#pragma unroll
    for (int off = 16; off > 0; off >>= 1) ls += __shfl_down(ls, off, 32);
    if (lane == 0) lred[wave] = ls;
    __syncthreads();
    if (tid == 0)
        partial[blockIdx.x] = lred[0] + lred[1] + lred[2] + lred[3];
}

// ---------------- kernel 3: finalize q_loss ----------------
__global__ __launch_bounds__(256) void vq_finalize(const float* __restrict__ partial,
                                                   float* __restrict__ loss_out) {
    __shared__ float s[256];
    float v = 0.f;
    for (int i = threadIdx.x; i < NTOT / MT; i += 256) v += partial[i];
    s[threadIdx.x] = v;
    __syncthreads();
    for (int o = 128; o > 0; o >>= 1) {
        if (threadIdx.x < o) s[threadIdx.x] += s[threadIdx.x + o];
        __syncthreads();
    }
    if (threadIdx.x == 0)
        loss_out[0] = 1.25f * s[0] / (float)((size_t)NTOT * CDIM);  // (1+BETA)*mean
}

extern "C" void kernel_launch(void* const* d_in, const int* in_sizes, int n_in,
                              void* d_out, int out_size, void* d_ws, size_t ws_size,
                              hipStream_t stream) {
    const float* z_e = (const float*)d_in[0];   // [16,128,32,32]
    const float* cb  = (const float*)d_in[1];   // [8192,128]
    float* out       = (float*)d_out;           // latent (2097152) + q_loss (1)
    float* latent    = out;
    float* loss      = out + (out_size - 1);

    float* ws_partial = (float*)d_ws;           // 1024 floats
    float* ws_cb2     = ws_partial + (NTOT / MT); // 8192 floats

    vq_cb_norms<<<KCB / 256, 256, 0, stream>>>(cb, ws_cb2);
    vq_main<<<NTOT / MT, 128, 0, stream>>>(z_e, cb, ws_cb2, latent, ws_partial);
    vq_finalize<<<1, 256, 0, stream>>>(ws_partial, loss);
}